// VolumeRenderer_19774029431716
// MI455X (gfx1250) — compile-verified
//
#include <hip/hip_runtime.h>

// Volume rendering (NeRF-style) for MI455X / gfx1250 — v2.
//
// Memory-bound: 340 MB traffic -> ~15us floor at 23.3 TB/s.
// exclusive cumprod(exp(-sigma*delta)) == exp(-prefix_sum(sigma*delta));
// the WITHIN-TILE (16-point) exclusive prefix is a matmul against a
// constant strict-lower-triangular ones matrix via V_WMMA_F32_16X16X4_F32
// (fp32 A/B -> no precision loss), and the inter-tile prefix is a per-ray
// scalar carry built from the B values each lane already holds.
//
// One wave = 16 rays x 64 points. Only sigma*delta lives in LDS (needed
// transposed for the WMMA B gather); features/depth are re-read from
// global in the epilogue where the lane pair (l, l^16) consumes each
// ray's contiguous rows straight out of L0.

#define NPTS      64
#define FAR_DELTA 1e10f
#define TILE_RAYS 16

typedef __attribute__((ext_vector_type(2))) float v2f;
typedef __attribute__((ext_vector_type(4))) float f4v;
typedef __attribute__((ext_vector_type(8))) float v8f;

__global__ __launch_bounds__(32) void volrend_wmma_kernel(
    const float* __restrict__ density,   // (N, 64)
    const float* __restrict__ depth,     // (N, 64)
    const float* __restrict__ feat,      // (N, 64, 3)
    float* __restrict__ feat_out,        // (N, 3)
    float* __restrict__ depth_out,       // (N)
    int n_rays)
{
    // sigma*delta, transposed [point][ray], ray-stride 17 (conflict-free).
    __shared__ float sdT[NPTS * 17];     // 4.3 KB -> 32 WGs/WGP occupancy cap

    const int lane = threadIdx.x;        // 0..31 (wave32)
    const int half = lane >> 4;          // 0: lanes 0-15, 1: lanes 16-31
    const int l16  = lane & 15;
    const int R0   = blockIdx.x * TILE_RAYS;

    // Warm L2 for this tile's streams (gfx1250 global_prefetch_b8).
    __builtin_prefetch(depth   + (size_t)R0 * NPTS,     0, 1);
    __builtin_prefetch(density + (size_t)R0 * NPTS,     0, 1);
    __builtin_prefetch(feat    + (size_t)R0 * NPTS * 3, 0, 1);

    // ---- Stage sigma*delta: coalesced b128, deltas in registers ----
    // 256 float4s; lane handles 8.  fidx -> (ray r, point p = 4*l16).
    #pragma unroll
    for (int i = 0; i < 8; ++i) {
        int f4i  = i * 32 + lane;
        int fidx = f4i * 4;
        int r    = fidx >> 6;                 // 0..15
        int p    = fidx & 63;                 // = 4*l16
        int row  = min(R0 + r, n_rays - 1);   // clamp (never hit for N=256K)
        f4v dv = *(const f4v*)(depth   + (size_t)row * NPTS + p);
        f4v sv = *(const f4v*)(density + (size_t)row * NPTS + p);
        // depth[p+4] lives in the next lane's dv.x (same ray for l16 < 15;
        // l16 == 15 is exactly the sentinel point p+3 == 63).
        float dnext = __shfl_down(dv.x, 1, 32);
        float dl3   = (l16 == 15) ? FAR_DELTA : (dnext - dv.w);
        sdT[(p + 0) * 17 + r] = sv.x * (dv.y - dv.x);
        sdT[(p + 1) * 17 + r] = sv.y * (dv.z - dv.y);
        sdT[(p + 2) * 17 + r] = sv.z * (dv.w - dv.z);
        sdT[(p + 3) * 17 + r] = sv.w * dl3;
    }
    __syncthreads();   // single-wave WG: near-NOP, orders LDS for the wave

    // ---- Constant strict-lower-triangular A masks (same for every tile) ----
    // A (16x4 f32) layout: lanes 0-15 hold K=0,1 ; lanes 16-31 hold K=2,3.
    v2f amask[4];
    #pragma unroll
    for (int q = 0; q < 4; ++q) {
        int j0 = 4 * q + half * 2;            // K row relative to tile base
        amask[q].x = (j0     < l16) ? 1.f : 0.f;
        amask[q].y = (j0 + 1 < l16) ? 1.f : 0.f;
    }

    // Per-ray global row pointers for the epilogue (lane = ray).
    const size_t rowc = (size_t)min(R0 + l16, n_rays - 1);
    const float* fr = feat  + rowc * NPTS * 3;
    const float* dr = depth + rowc * NPTS;

    float carry = 0.f;                        // prefix of tiles < t (per ray)
    float accx = 0.f, accy = 0.f, accz = 0.f, accd = 0.f;

    #pragma unroll
    for (int t = 0; t < 4; ++t) {             // tile: points 16t..16t+15
        v8f   C = {};
        float tpart = 0.f;                    // this lane's share of tile sum
        #pragma unroll
        for (int q = 0; q < 4; ++q) {
            int j0 = 16 * t + 4 * q + half * 2;   // B K rows (f32 layout)
            v2f b;
            b.x = sdT[(j0    ) * 17 + l16];       // B col = ray
            b.y = sdT[(j0 + 1) * 17 + l16];
            tpart += b.x + b.y;
            C = __builtin_amdgcn_wmma_f32_16x16x4_f32(
                    false, amask[q], false, b, (short)0, C, false, false);
        }
        // D layout: lane = ray; VGPR v -> point 16t + v (+8 for upper half).
        #pragma unroll
        for (int v = 0; v < 8; ++v) {
            int   p    = 16 * t + v + half * 8;
            float Pexc = carry + C[v];            // sum_{j<p} sigma*delta_j
            float sdp  = sdT[p * 17 + l16];
            float w    = __expf(-Pexc) - __expf(-(Pexc + sdp)); // T*(1-e)
            accx += w * fr[p * 3 + 0];
            accy += w * fr[p * 3 + 1];
            accz += w * fr[p * 3 + 2];
            accd += w * dr[p];
        }
        carry += tpart + __shfl_xor(tpart, 16, 32);   // full tile sum per ray
    }

    // Merge the two half-wave point ranges (ray lives in lanes l and l^16).
    accx += __shfl_xor(accx, 16, 32);
    accy += __shfl_xor(accy, 16, 32);
    accz += __shfl_xor(accz, 16, 32);
    accd += __shfl_xor(accd, 16, 32);

    if (lane < 16) {
        int rg = R0 + l16;
        if (rg < n_rays) {
            feat_out[rg * 3 + 0] = accx;
            feat_out[rg * 3 + 1] = accy;
            feat_out[rg * 3 + 2] = accz;
            depth_out[rg]        = accd;
        }
    }
}

extern "C" void kernel_launch(void* const* d_in, const int* in_sizes, int n_in,
                              void* d_out, int out_size, void* d_ws, size_t ws_size,
                              hipStream_t stream) {
    (void)n_in; (void)out_size; (void)d_ws; (void)ws_size;
    const float* density = (const float*)d_in[0];   // (N,64,1)
    const float* depthv  = (const float*)d_in[1];   // (N,64)
    const float* feat    = (const float*)d_in[2];   // (N,64,3)

    const int n_rays = in_sizes[1] / NPTS;          // from depth_values
    float* feat_out  = (float*)d_out;               // (N,3) flat first
    float* depth_out = feat_out + (size_t)n_rays * 3;

    const int tiles = (n_rays + TILE_RAYS - 1) / TILE_RAYS;
    volrend_wmma_kernel<<<dim3(tiles), dim3(32), 0, stream>>>(
        density, depthv, feat, feat_out, depth_out, n_rays);
}